// BSpline1D_46385646796839
// MI455X (gfx1250) — compile-verified
//
#include <hip/hip_runtime.h>

typedef float v2f __attribute__((ext_vector_type(2)));
typedef float v8f __attribute__((ext_vector_type(8)));
typedef float f4v __attribute__((ext_vector_type(4)));

#define NB    16   // n_basis
#define DEG   3
#define NKNOT 20   // (DEG+1)*2 + 12 interior
#define NSPANS 13  // NB - DEG

__device__ __forceinline__ float safe_inv(float d) {
    return (fabsf(d) > 1e-12f) ? (1.0f / d) : 0.0f;
}

// ---------------------------------------------------------------------------
// Setup kernel: one wave computes the 4 boundary spline values via WMMA:
//   D(16x16) = A(16x4 chunks of dense basis rows) x B(coeffs broadcast) + C
// Rows 0..3 of A are the dense 16-vector basis at x = {0, 0.001, 1, 0.999};
// column-broadcast B gives D[m][*] = S(x_m). Writes {y0_lo, slope_lo,
// y0_hi, slope_hi} to cons[0..3].
// ---------------------------------------------------------------------------
__global__ void __launch_bounds__(32)
bspline_setup_kernel(const float* __restrict__ coeffs,
                     const float* __restrict__ knots,
                     float* __restrict__ cons) {
    __shared__ float kn[NKNOT];
    __shared__ float cf[NB];
    int t = threadIdx.x;
    if (t < NKNOT) kn[t] = knots[t];
    if (t < NB)    cf[t] = coeffs[t];
    __syncthreads();

    int lane = t & 31;
    int m    = lane & 15;                 // matrix row this lane feeds
    float xq = (m == 0) ? 0.0f : (m == 1) ? 0.001f : (m == 2) ? 1.0f : 0.999f;
    float rowmask = (m < 4) ? 1.0f : 0.0f;   // rows 4..15 are zero

    // Dense Cox-de Boor recursion, exactly mirroring the reference
    float B[NB];
#pragma unroll
    for (int k = 0; k < NB; ++k) {
        bool inb = (k == NB - 1) ? (xq >= kn[k] && xq <= kn[k + 1])
                                 : (xq >= kn[k] && xq <  kn[k + 1]);
        B[k] = inb ? rowmask : 0.0f;
    }
#pragma unroll
    for (int p = 1; p <= DEG; ++p) {
#pragma unroll
        for (int k = 0; k < NB; ++k) {       // ascending: B[k+1] still level p-1
            float i1 = safe_inv(kn[k + p] - kn[k]);
            float i2 = safe_inv(kn[k + p + 1] - kn[k + 1]);
            float Bn = (k < NB - 1) ? B[k + 1] : 0.0f;
            B[k] = (xq - kn[k]) * i1 * B[k] + (kn[k + p + 1] - xq) * i2 * Bn;
        }
    }

    // Chain 4 x V_WMMA_F32_16X16X4_F32 over K = 0..15.
    // A 16x4 f32 layout: lanes 0-15 = {K0,K1}, lanes 16-31 = {K2,K3} (same M).
    // B 4x16 f32 layout mirrors C/D row striping: VGPR0 = rows {K0,K2},
    // VGPR1 = rows {K1,K3}; value depends only on K (column broadcast).
    bool hi = lane >= 16;
    v8f acc = {0.f, 0.f, 0.f, 0.f, 0.f, 0.f, 0.f, 0.f};
#pragma unroll
    for (int j = 0; j < 4; ++j) {
        v2f a, b;
        a.x = hi ? B[4 * j + 2]  : B[4 * j + 0];
        a.y = hi ? B[4 * j + 3]  : B[4 * j + 1];
        b.x = hi ? cf[4 * j + 2] : cf[4 * j + 0];
        b.y = hi ? cf[4 * j + 3] : cf[4 * j + 1];
        acc = __builtin_amdgcn_wmma_f32_16x16x4_f32(
                  false, a, false, b, (short)0, acc, false, false);
    }

    // D layout: VGPR v, lanes 0-15 hold row M=v -> acc[m] on lane 0 is S(x_m).
    if (t == 0) {
        float y0_lo = acc[0];
        float y1_lo = acc[1];
        float y0_hi = acc[2];
        float y1_hi = acc[3];
        cons[0] = y0_lo;
        cons[1] = (y1_lo - y0_lo) / 0.001f;   // slope_lo
        cons[2] = y0_hi;
        cons[3] = (y0_hi - y1_hi) / 0.001f;   // slope_hi
    }
}

// ---------------------------------------------------------------------------
// Per-element evaluation: de Boor on the 4-wide local support.
// tab[j] = (t_j, 1/(t_{j+1}-t_j), 1/(t_{j+2}-t_j), 1/(t_{j+3}-t_j))
// ---------------------------------------------------------------------------
__device__ __forceinline__ float
eval_one(float xv, const float4* tab, const float* cf,
         float y0_lo, float slope_lo, float y0_hi, float slope_hi) {
    float xin = fminf(fmaxf(xv, 0.0f), 1.0f);
    int sl = (int)(xin * (float)NSPANS);
    sl = (sl > NSPANS - 1) ? NSPANS - 1 : sl;
    int s = sl + DEG;                      // knot-span index, 3..15

    float4 t2 = tab[s - 2];
    float4 t1 = tab[s - 1];
    float4 t0 = tab[s];
    float d0 = cf[s - 3], d1 = cf[s - 2], d2 = cf[s - 1], d3 = cf[s];

    float dx2 = xin - t2.x, dx1 = xin - t1.x, dx0 = xin - t0.x;
    // level r=1 (denominators t_{i+3}-t_i)
    float e1 = fmaf(dx2 * t2.w, d1 - d0, d0);
    float e2 = fmaf(dx1 * t1.w, d2 - d1, d1);
    float e3 = fmaf(dx0 * t0.w, d3 - d2, d2);
    // level r=2 (denominators t_{i+2}-t_i)
    float f2 = fmaf(dx1 * t1.z, e2 - e1, e1);
    float f3 = fmaf(dx0 * t0.z, e3 - e2, e2);
    // level r=3 (denominator t_{i+1}-t_i)
    float S  = fmaf(dx0 * t0.y, f3 - f2, f2);

    float y = (xv < 0.0f) ? fmaf(slope_lo, xv, y0_lo) : S;
    y = (xv > 1.0f) ? fmaf(slope_hi, xv - 1.0f, y0_hi) : y;
    return y;
}

// ---------------------------------------------------------------------------
// Streaming kernel: 128-bit loads, prefetch one grid stride ahead,
// non-temporal 128-bit stores. Memory bound: 67 MB @ 23.3 TB/s ~ 2.9 us.
// ---------------------------------------------------------------------------
__global__ void __launch_bounds__(256)
bspline_main_kernel(const float* __restrict__ x,
                    const float* __restrict__ coeffs,
                    const float* __restrict__ knots,
                    const float* __restrict__ cons,
                    float* __restrict__ out, int n) {
    __shared__ float  kn[NKNOT];
    __shared__ float  cf[NB];
    __shared__ float4 tab[NB];

    int t = threadIdx.x;
    if (t < NKNOT) kn[t] = knots[t];
    if (t < NB)    cf[t] = coeffs[t];
    __syncthreads();
    if (t < NB) {
        float tj = kn[t];
        tab[t] = make_float4(tj,
                             safe_inv(kn[t + 1] - tj),
                             safe_inv(kn[t + 2] - tj),
                             safe_inv(kn[t + 3] - tj));
    }
    __syncthreads();

    float4 cv = *(const float4*)cons;
    float y0_lo = cv.x, slope_lo = cv.y, y0_hi = cv.z, slope_hi = cv.w;

    int gtid   = blockIdx.x * blockDim.x + t;
    int stride = gridDim.x * blockDim.x;
    int n4 = n >> 2;
    const f4v* x4 = (const f4v*)x;
    f4v* o4 = (f4v*)out;

    for (int i = gtid; i < n4; i += stride) {
        __builtin_prefetch(&x4[i + stride], 0, 0);   // global_prefetch_b8
        f4v v = x4[i];
        f4v r;
        r.x = eval_one(v.x, tab, cf, y0_lo, slope_lo, y0_hi, slope_hi);
        r.y = eval_one(v.y, tab, cf, y0_lo, slope_lo, y0_hi, slope_hi);
        r.z = eval_one(v.z, tab, cf, y0_lo, slope_lo, y0_hi, slope_hi);
        r.w = eval_one(v.w, tab, cf, y0_lo, slope_lo, y0_hi, slope_hi);
        __builtin_nontemporal_store(r, &o4[i]);      // write-once output
    }
    // scalar tail (n % 4), empty for n = 8388608
    for (int i = (n4 << 2) + gtid; i < n; i += stride) {
        out[i] = eval_one(x[i], tab, cf, y0_lo, slope_lo, y0_hi, slope_hi);
    }
}

// ---------------------------------------------------------------------------
extern "C" void kernel_launch(void* const* d_in, const int* in_sizes, int n_in,
                              void* d_out, int out_size, void* d_ws, size_t ws_size,
                              hipStream_t stream) {
    const float* x      = (const float*)d_in[0];
    const float* coeffs = (const float*)d_in[1];
    const float* knots  = (const float*)d_in[2];
    float* out  = (float*)d_out;
    float* cons = (float*)d_ws;        // 4 floats of scratch for boundary consts
    int n = in_sizes[0];

    bspline_setup_kernel<<<1, 32, 0, stream>>>(coeffs, knots, cons);

    int n4 = n >> 2;
    int blocks = (n4 + 255) / 256;
    if (blocks > 2048) blocks = 2048;
    if (blocks < 1)    blocks = 1;
    bspline_main_kernel<<<blocks, 256, 0, stream>>>(x, coeffs, knots, cons, out, n);
}